// RobustPINN_77910706749677
// MI455X (gfx1250) — compile-verified
//
#include <hip/hip_runtime.h>
#include <hip/hip_bf16.h>
#include <math.h>

// CDNA5 / gfx1250: wave32, fp32 WMMA 16x16x4.
typedef __attribute__((ext_vector_type(2))) float v2f;
typedef __attribute__((ext_vector_type(8))) float v8f;

#define WIDTH   50
#define KPAD    52      // 13 K-tiles of 4 (50 -> 52, pad rows are zero)
#define NPAD    64      // 4 N-tiles of 16 (50 -> 64, pad cols are zero)
#define NCH     6       // jet channels: val, d/dx, d2/dx2, d3/dx3, d4/dx4, d/dt
#define PTSW    32      // points per wave (2 WMMA M-tiles of 16)
#define EPSF    1e-5f

// LDS budget (single wave per block, all static):
//   weights  : KPAD*NPAD floats = 13,312 B  (restaged per hidden layer)
//   act jets : NCH*PTSW*NPAD    = 49,152 B
//   total 62,464 B < 64 KB
#define W_FLOATS (KPAD * NPAD)
#define H_FLOATS (NCH * PTSW * NPAD)

// LayerNorm + tanh forward-jet, applied in place. Lane `lane` owns point row
// `lane`. Channels: a0=value, a1..a4 = d^k/dx^k, a5 = d/dt. All derivatives
// are total derivatives in x (t held fixed), so the chain rules below are the
// exact Faa-di-Bruno compositions the nested jax.grad computes.
__device__ __forceinline__ void ln_tanh_jet(float* __restrict__ Hbuf, int lane,
                                            const float* __restrict__ g,
                                            const float* __restrict__ b) {
  float* z0 = Hbuf + (0 * PTSW + lane) * NPAD;
  float* z1 = Hbuf + (1 * PTSW + lane) * NPAD;
  float* z2 = Hbuf + (2 * PTSW + lane) * NPAD;
  float* z3 = Hbuf + (3 * PTSW + lane) * NPAD;
  float* z4 = Hbuf + (4 * PTSW + lane) * NPAD;
  float* z5 = Hbuf + (5 * PTSW + lane) * NPAD;

  const float inv = 1.0f / (float)WIDTH;

  // --- means of every jet channel (mean is linear => jets of mu) ---
  float m0 = 0.f, m1 = 0.f, m2 = 0.f, m3 = 0.f, m4 = 0.f, m5 = 0.f;
  for (int j = 0; j < WIDTH; ++j) {
    m0 += z0[j]; m1 += z1[j]; m2 += z2[j];
    m3 += z3[j]; m4 += z4[j]; m5 += z5[j];
  }
  m0 *= inv; m1 *= inv; m2 *= inv; m3 *= inv; m4 *= inv; m5 *= inv;

  // --- jets of s = var + eps, var = mean(a0^2), a_k = z_k - m_k ---
  float p00 = 0.f, p01 = 0.f, p11 = 0.f, p02 = 0.f, p12 = 0.f;
  float p03 = 0.f, p22 = 0.f, p13 = 0.f, p04 = 0.f, p05 = 0.f;
  for (int j = 0; j < WIDTH; ++j) {
    float a0 = z0[j] - m0, a1 = z1[j] - m1, a2 = z2[j] - m2;
    float a3 = z3[j] - m3, a4 = z4[j] - m4, a5 = z5[j] - m5;
    p00 += a0 * a0; p01 += a0 * a1; p11 += a1 * a1;
    p02 += a0 * a2; p12 += a1 * a2; p03 += a0 * a3;
    p22 += a2 * a2; p13 += a1 * a3; p04 += a0 * a4;
    p05 += a0 * a5;
  }
  float s  = p00 * inv + EPSF;
  float s1 = 2.f * p01 * inv;
  float s2 = 2.f * (p11 + p02) * inv;
  float s3 = 2.f * (3.f * p12 + p03) * inv;
  float s4 = 2.f * (3.f * p22 + 4.f * p13 + p04) * inv;
  float s5 = 2.f * p05 * inv;

  // --- jets of r = s^(-1/2) ---
  float r  = 1.0f / sqrtf(s);
  float is = 1.0f / s;
  float f1 = -0.5f     * r * is;             // d r/ds
  float f2 =  0.75f    * r * is * is;
  float f3 = -1.875f   * r * is * is * is;
  float f4 =  6.5625f  * r * is * is * is * is;
  float r1 = f1 * s1;
  float r2 = f2 * s1 * s1 + f1 * s2;
  float r3 = f3 * s1 * s1 * s1 + 3.f * f2 * s1 * s2 + f1 * s3;
  float r4 = f4 * s1 * s1 * s1 * s1 + 6.f * f3 * s1 * s1 * s2
           + f2 * (3.f * s2 * s2 + 4.f * s1 * s3) + f1 * s4;
  float r5 = f1 * s5;

  // --- per-neuron: y = g*a0*r + b (Leibniz), then tanh jet ---
  for (int j = 0; j < WIDTH; ++j) {
    float a0 = z0[j] - m0, a1 = z1[j] - m1, a2 = z2[j] - m2;
    float a3 = z3[j] - m3, a4 = z4[j] - m4, a5 = z5[j] - m5;
    float gg = g[j], bb = b[j];
    float y0 = gg * (a0 * r) + bb;
    float y1 = gg * (a1 * r + a0 * r1);
    float y2 = gg * (a2 * r + 2.f * a1 * r1 + a0 * r2);
    float y3 = gg * (a3 * r + 3.f * a2 * r1 + 3.f * a1 * r2 + a0 * r3);
    float y4 = gg * (a4 * r + 4.f * a3 * r1 + 6.f * a2 * r2
                     + 4.f * a1 * r3 + a0 * r4);
    float y5 = gg * (a5 * r + a0 * r5);

    float T  = tanhf(y0);
    float pm = 1.f - T * T;                       // tanh'
    float d2 = -2.f * T * pm;                     // tanh''
    float d3 = -2.f * pm * (1.f - 3.f * T * T);   // tanh'''
    float d4 = 8.f * T * pm * (2.f - 3.f * T * T);// tanh''''

    z0[j] = T;
    z1[j] = pm * y1;
    z2[j] = d2 * y1 * y1 + pm * y2;
    z3[j] = d3 * y1 * y1 * y1 + 3.f * d2 * y1 * y2 + pm * y3;
    z4[j] = d4 * y1 * y1 * y1 * y1 + 6.f * d3 * y1 * y1 * y2
          + d2 * (3.f * y2 * y2 + 4.f * y1 * y3) + pm * y4;
    z5[j] = pm * y5;
  }
}

__global__ __launch_bounds__(32) void pinn_jet_kernel(
    const float* __restrict__ x,   const float* __restrict__ t,
    const float* __restrict__ lb,  const float* __restrict__ ub,
    const float* __restrict__ W0,  const float* __restrict__ b0,
    const float* __restrict__ Wh,  const float* __restrict__ bh,
    const float* __restrict__ Wl,  const float* __restrict__ bl,
    const float* __restrict__ lng, const float* __restrict__ lnb,
    float* __restrict__ out) {
  __shared__ float Wt[W_FLOATS];   // Wt[k][j] = Wh[l][j][k], zero-padded
  __shared__ float Hbuf[H_FLOATS]; // Hbuf[c][row][col], row = lane's point

  const int lane = threadIdx.x;          // single wave32 per block
  const int pt   = blockIdx.x * PTSW + lane;

  // zero padding columns 50..63 once (WMMA K-tile 12 reads cols 48..51;
  // GEMM writeback keeps cols >=50 at zero because weight pad rows/cols are 0)
  for (int c = 0; c < NCH; ++c)
    for (int col = WIDTH; col < NPAD; ++col)
      Hbuf[(c * PTSW + lane) * NPAD + col] = 0.f;

  // ---------------- input scaling + layer 0 (K=2, cheap VALU) -------------
  {
    float lb0 = lb[0], lb1 = lb[1], ub0 = ub[0], ub1 = ub[1];
    float hx = -1.f + 2.f * (x[pt] - lb0) / (ub0 - lb0);
    float ht = -1.f + 2.f * (t[pt] - lb1) / (ub1 - lb1);
    float sx = 2.f / (ub0 - lb0);          // d(hx)/dx
    float st = 2.f / (ub1 - lb1);          // d(ht)/dt
    float* h0 = Hbuf + (0 * PTSW + lane) * NPAD;
    float* h1 = Hbuf + (1 * PTSW + lane) * NPAD;
    float* h2 = Hbuf + (2 * PTSW + lane) * NPAD;
    float* h3 = Hbuf + (3 * PTSW + lane) * NPAD;
    float* h4 = Hbuf + (4 * PTSW + lane) * NPAD;
    float* h5 = Hbuf + (5 * PTSW + lane) * NPAD;
    for (int j = 0; j < WIDTH; ++j) {
      float w0 = W0[2 * j], w1 = W0[2 * j + 1];
      h0[j] = w0 * hx + w1 * ht + b0[j];
      h1[j] = w0 * sx;                     // d/dx seed
      h2[j] = 0.f; h3[j] = 0.f; h4[j] = 0.f;
      h5[j] = w1 * st;                     // d/dt seed
    }
    ln_tanh_jet(Hbuf, lane, lng, lnb);
  }

  const int rlo = lane & 15;     // M/N position within a 16-wide tile
  const int hb  = lane >> 4;     // lane half selects K pair / row offset (ISA 7.12.2)

  // ---------------- 4 hidden layers: fp32 WMMA GEMM + jet nonlinearity ----
  for (int l = 0; l < 4; ++l) {
    // stage Wt[k][j] = Wh[l][j][k] (transposed, zero-padded) into LDS
    const float* Wg = Wh + l * WIDTH * WIDTH;
    for (int idx = lane; idx < W_FLOATS; idx += 32) {
      int k = idx / NPAD, j = idx - k * NPAD;
      Wt[idx] = (k < WIDTH && j < WIDTH) ? Wg[j * WIDTH + k] : 0.f;
    }
    // single wave: in-wave DS ordering makes the staged weights visible;
    // no barrier needed, EXEC stays all-ones for the WMMA block below.

    for (int m = 0; m < 2; ++m) {          // two 16-point M-tiles
      const int arow = m * 16 + rlo;
      for (int c = 0; c < NCH; ++c) {      // jet channels share W
        float* Hc = Hbuf + c * PTSW * NPAD;
        // hoist all 13 A fragments so we can write D back over Hc in place.
        // A 16x4 f32 layout: lanes 0-15 K={0,1}, lanes 16-31 K={2,3}.
        v2f a[13];
#pragma unroll
        for (int k = 0; k < 13; ++k)
          a[k] = *(const v2f*)&Hc[arow * NPAD + 4 * k + 2 * hb];
#pragma unroll
        for (int n = 0; n < 4; ++n) {      // four 16-col N-tiles
          const int bcol = n * 16 + rlo;
          v8f acc = {};
#pragma unroll
          for (int k = 0; k < 13; ++k) {
            // B 4x16 f32: VGPR0 rows {K0,K2}, VGPR1 rows {K1,K3} across halves
            v2f bb;
            bb.x = Wt[(4 * k + 2 * hb)     * NPAD + bcol];
            bb.y = Wt[(4 * k + 2 * hb + 1) * NPAD + bcol];
            acc = __builtin_amdgcn_wmma_f32_16x16x4_f32(
                false, a[k], false, bb, (short)0, acc, false, false);
          }
          // D layout: VGPR r -> row r (lanes 0-15) / r+8 (lanes 16-31)
          float bias = (c == 0 && bcol < WIDTH) ? bh[l * WIDTH + bcol] : 0.f;
#pragma unroll
          for (int r = 0; r < 8; ++r)
            Hc[(m * 16 + r + 8 * hb) * NPAD + bcol] = acc[r] + bias;
        }
      }
    }
    ln_tanh_jet(Hbuf, lane, lng + (l + 1) * WIDTH, lnb + (l + 1) * WIDTH);
  }

  // ---------------- final linear (1x50) + output [N,6] --------------------
  {
    float u0 = 0.f, u1 = 0.f, u2 = 0.f, u3 = 0.f, u4 = 0.f, u5 = 0.f;
    const float* h0 = Hbuf + (0 * PTSW + lane) * NPAD;
    const float* h1 = Hbuf + (1 * PTSW + lane) * NPAD;
    const float* h2 = Hbuf + (2 * PTSW + lane) * NPAD;
    const float* h3 = Hbuf + (3 * PTSW + lane) * NPAD;
    const float* h4 = Hbuf + (4 * PTSW + lane) * NPAD;
    const float* h5 = Hbuf + (5 * PTSW + lane) * NPAD;
    for (int j = 0; j < WIDTH; ++j) {
      float w = Wl[j];
      u0 += w * h0[j]; u1 += w * h1[j]; u2 += w * h2[j];
      u3 += w * h3[j]; u4 += w * h4[j]; u5 += w * h5[j];
    }
    float* o = out + (size_t)pt * 6;
    o[0] = u0 + bl[0]; o[1] = u1; o[2] = u2;
    o[3] = u3; o[4] = u4; o[5] = u5;
  }
}

extern "C" void kernel_launch(void* const* d_in, const int* in_sizes, int n_in,
                              void* d_out, int out_size, void* d_ws, size_t ws_size,
                              hipStream_t stream) {
  (void)n_in; (void)out_size; (void)d_ws; (void)ws_size;
  const float* x   = (const float*)d_in[0];
  const float* t   = (const float*)d_in[1];
  const float* lb  = (const float*)d_in[2];
  const float* ub  = (const float*)d_in[3];
  const float* W0  = (const float*)d_in[4];
  const float* b0  = (const float*)d_in[5];
  const float* Wh  = (const float*)d_in[6];
  const float* bh  = (const float*)d_in[7];
  const float* Wl  = (const float*)d_in[8];
  const float* bl  = (const float*)d_in[9];
  const float* lng = (const float*)d_in[10];
  const float* lnb = (const float*)d_in[11];

  const int npts   = in_sizes[0];          // 131072, divisible by 32
  const int blocks = npts / PTSW;
  hipLaunchKernelGGL(pinn_jet_kernel, dim3(blocks), dim3(PTSW), 0, stream,
                     x, t, lb, ub, W0, b0, Wh, bh, Wl, bl, lng, lnb,
                     (float*)d_out);
}